// KalmanFilter_71786083385553
// MI455X (gfx1250) — compile-verified
//
#include <hip/hip_runtime.h>

// Problem constants (from reference)
#define B_N 1024
#define T_N 200
#define S_N 24
#define M_N 2
#define EPS 1e-3f

// Flat float offsets inside d_out (return order: means, covs, Rs, Hs)
#define MEANS_OFF 0
#define COVS_OFF  4915200      // B*T*S
#define RS_OFF    122880000    // + B*T*S*S
#define HS_OFF    123699200    // + B*T*M*M

typedef __attribute__((ext_vector_type(2))) float v2f;
typedef __attribute__((ext_vector_type(4))) float v4f;
typedef __attribute__((ext_vector_type(8))) float v8f;

// ---------------------------------------------------------------------------
// Kernel A: single workgroup computes the data-independent covariance
// sequence cov_t, plus the mean-recurrence matrices A_t = F(I-K_t H) and
// Bm_t = F K_t, for t = 1..T-1.  576 threads = one per (i,j) of a 24x24.
// wA is padded to [T][32][24] (rows 24..31 zero) and wBT is [T][2][32][2]
// (Bm_t^T pairs, zero outside half 0 / n<24) so kernel C loads operands
// unconditionally -- no exec-mask branches in its serial loop.
// ---------------------------------------------------------------------------
__global__ __launch_bounds__(576) void kf_cov_seq(
    const float* __restrict__ Fg, const float* __restrict__ Hg,
    const float* __restrict__ Lq, const float* __restrict__ Lr,
    float* __restrict__ wcov,   // [T][24][24]
    float* __restrict__ wA,     // [T][32][24]  (slot t=0 unused, rows 24-31 = 0)
    float* __restrict__ wBT)    // [T][2][32][2] (slot t=0 unused, padded w/ 0)
{
    __shared__ float Fsh[24][25], Qsh[24][25];
    __shared__ float cov[24][25], P[24][25];
    __shared__ float Hsh[2][24], Hcov[2][24];
    __shared__ float covHT[24][2], Ksh[24][2], FK[24][2];
    __shared__ float Sm[2][2], Rsh[2][2];

    const int tid = threadIdx.x;
    const int i = tid / 24, j = tid % 24;

    // ---- init: F, H, Q = Lq Lq^T + eps I, R = Lr Lr^T + eps I, cov = I ----
    Fsh[i][j] = Fg[i * 24 + j];
    if (tid < 48) Hsh[tid / 24][tid % 24] = Hg[tid];
    if (tid < 4) {
        int m = tid >> 1, n = tid & 1;
        Rsh[m][n] = Lr[m * 2 + 0] * Lr[n * 2 + 0]
                  + Lr[m * 2 + 1] * Lr[n * 2 + 1]
                  + (m == n ? EPS : 0.0f);
    }
    float q = 0.0f;
    #pragma unroll
    for (int k = 0; k < 24; ++k) q += Lq[i * 24 + k] * Lq[j * 24 + k];
    Qsh[i][j] = q + (i == j ? EPS : 0.0f);
    float id = (i == j) ? 1.0f : 0.0f;
    cov[i][j] = id;
    wcov[i * 24 + j] = id;          // cov_0 = I

    // zero the constant pad regions once (rows 24..31 of every wA slot,
    // and the whole wBT array; live entries are overwritten each step)
    for (int idx = tid; idx < T_N * 192; idx += 576)
        wA[(idx / 192) * 768 + 576 + (idx % 192)] = 0.0f;
    for (int idx = tid; idx < T_N * 128; idx += 576)
        wBT[idx] = 0.0f;
    __syncthreads();

    for (int t = 1; t < T_N; ++t) {
        // phase 1: covHT = cov H^T  and  Hcov = H cov
        if (tid < 48) {
            int s = tid % 24, m = tid / 24;
            float a = 0.0f;
            #pragma unroll
            for (int k = 0; k < 24; ++k) a += cov[s][k] * Hsh[m][k];
            covHT[s][m] = a;
        } else if (tid < 96) {
            int jj = (tid - 48) % 24, m = (tid - 48) / 24;
            float a = 0.0f;
            #pragma unroll
            for (int s = 0; s < 24; ++s) a += Hsh[m][s] * cov[s][jj];
            Hcov[m][jj] = a;
        }
        __syncthreads();

        // phase 2: innovation covariance S = H covHT + R  (2x2)
        if (tid < 4) {
            int m = tid >> 1, n = tid & 1;
            float a = 0.0f;
            #pragma unroll
            for (int s = 0; s < 24; ++s) a += Hsh[m][s] * covHT[s][n];
            Sm[m][n] = a + Rsh[m][n];
        }
        __syncthreads();

        // phase 3: K = covHT S^{-1}  (closed-form 2x2 inverse, per-thread)
        if (tid < 48) {
            int s = tid % 24, m = tid / 24;
            float det = Sm[0][0] * Sm[1][1] - Sm[0][1] * Sm[1][0];
            float inv = 1.0f / det;
            float si0 = inv * ((m == 0) ?  Sm[1][1] : -Sm[0][1]);
            float si1 = inv * ((m == 0) ? -Sm[1][0] :  Sm[0][0]);
            Ksh[s][m] = covHT[s][0] * si0 + covHT[s][1] * si1;
        }
        __syncthreads();

        // phase 4: cov' = cov - K (H cov)  (rank-2, in place) ;  FK = F K
        cov[i][j] = cov[i][j] - Ksh[i][0] * Hcov[0][j] - Ksh[i][1] * Hcov[1][j];
        if (tid < 48) {
            int s = tid % 24, m = tid / 24;
            float a = 0.0f;
            #pragma unroll
            for (int k = 0; k < 24; ++k) a += Fsh[s][k] * Ksh[k][m];
            FK[s][m] = a;
        }
        __syncthreads();

        // phase 5: P = F cov' ;  A_t = F - FK H ;  Bm_t^T = FK^T
        {
            float p = 0.0f;
            #pragma unroll
            for (int k = 0; k < 24; ++k) p += Fsh[i][k] * cov[k][j];
            P[i][j] = p;
            float a = Fsh[i][j] - FK[i][0] * Hsh[0][j] - FK[i][1] * Hsh[1][j];
            wA[t * 768 + i * 24 + j] = a;
            if (tid < 48) {
                int s = tid % 24, m = tid / 24;
                // wBT[t][0][s][m] = Bm_t[s][m]   (half 1 & s>=24 stay zero)
                wBT[t * 128 + s * 2 + m] = FK[s][m];
            }
        }
        __syncthreads();

        // phase 6: cov_t = P F^T + Q
        {
            float c = Qsh[i][j];
            #pragma unroll
            for (int k = 0; k < 24; ++k) c += P[i][k] * Fsh[j][k];
            cov[i][j] = c;
            wcov[t * 576 + i * 24 + j] = c;
        }
        __syncthreads();
    }
}

// ---------------------------------------------------------------------------
// Kernel C: batched mean recurrence via V_WMMA_F32_16X16X4_F32.
// One wave per 16 batches. Mt[16x24] <- Mt*A_t^T + Obs*Bm_t^T.
// A operand (16x4 f32, 2 VGPRs): lanes 0-15 hold K={k0,k0+1}, lanes 16-31
// hold K={k0+2,k0+3}, row M = lane%16. B operand (4x16): lane halves split K.
// C/D (16x16, 8 VGPRs): VGPR r -> row r (+8 for upper lanes), col = lane%16.
// All operand loads are unconditional (workspace is zero-padded).
// ---------------------------------------------------------------------------
__global__ __launch_bounds__(32) void kf_means(
    const float* __restrict__ obs,   // [B][T][M]
    const float* __restrict__ wA,    // [T][32][24], rows 24-31 zero
    const float* __restrict__ wBT,   // [T][2][32][2], zero-padded
    float* __restrict__ means)       // [B][T][S]
{
    __shared__ float Mt[16][33];     // 33-stride: conflict-free column access

    const int l    = threadIdx.x;
    const int lm   = l & 15;
    const int half = l >> 4;
    const int b0   = blockIdx.x * 16;

    // mean_0 = 0 : init LDS state and write means[:,0,:]
    for (int idx = l; idx < 16 * S_N; idx += 32) {
        int m = idx / S_N, s = idx % S_N;
        Mt[m][s] = 0.0f;
        means[(size_t)(b0 + m) * T_N * S_N + s] = 0.0f;
    }
    __syncthreads();

    for (int t = 1; t < T_N; ++t) {
        // --- A operand: current means tile, 6 chunks of K=4 (states 0..23)
        v2f a[6];
        #pragma unroll
        for (int kk = 0; kk < 6; ++kk) {
            int k = kk * 4 + 2 * half;
            a[kk].x = Mt[lm][k];
            a[kk].y = Mt[lm][k + 1];
        }
        // --- A operand for obs part: 16x2 (K rows 2,3 are zero padding).
        // All lanes load (upper half duplicates lanes 0-15), then select 0.
        v2f oa;
        {
            size_t off = ((size_t)(b0 + lm) * T_N + (t - 1)) * M_N;
            float ox = obs[off];
            float oy = obs[off + 1];
            oa.x = half ? 0.0f : ox;
            oa.y = half ? 0.0f : oy;
        }

        #pragma unroll
        for (int nt = 0; nt < 2; ++nt) {
            int n = nt * 16 + lm;            // output state column (pad >= 24)
            bool valid = (n < S_N);
            v8f acc = {};

            // obs contribution: B = Bm_t^T (zero-padded table, b64 load)
            v2f bb;
            {
                const float* bp = wBT + t * 128 + half * 64 + n * 2;
                bb.x = bp[0];
                bb.y = bp[1];
            }
            acc = __builtin_amdgcn_wmma_f32_16x16x4_f32(
                false, oa, false, bb, (short)0, acc, false, false);

            // mean contribution: B = A_t^T, K = 0..23 in chunks of 4
            #pragma unroll
            for (int kk = 0; kk < 6; ++kk) {
                int k = kk * 4 + 2 * half;
                const float* ap = wA + t * 768 + n * 24 + k;
                v2f bm;
                bm.x = ap[0];
                bm.y = ap[1];
                acc = __builtin_amdgcn_wmma_f32_16x16x4_f32(
                    false, a[kk], false, bm, (short)0, acc, false, false);
            }

            // write means[b, t, n] and stage next-step state into LDS
            #pragma unroll
            for (int r = 0; r < 8; ++r) {
                int m = r + 8 * half;
                if (valid) {
                    means[(size_t)(b0 + m) * T_N * S_N + (size_t)t * S_N + n] =
                        acc[r];
                    Mt[m][n] = acc[r];
                }
            }
        }
        __syncthreads();
    }
}

// ---------------------------------------------------------------------------
// Broadcast kernels: pure write-bandwidth, compile-time modulo period,
// non-temporal 128-bit stores (output never re-read; keep L2 for the source).
// ---------------------------------------------------------------------------
template <int PERIOD4>
__global__ void kf_broadcast4(const v4f* __restrict__ src,
                              v4f* __restrict__ dst, int n4)
{
    int idx = blockIdx.x * blockDim.x + threadIdx.x;
    if (idx < n4) {
        v4f v = src[idx % PERIOD4];
        __builtin_nontemporal_store(v, &dst[idx]);
    }
}

__global__ void kf_broadcast_R(const float* __restrict__ Lr,
                               v4f* __restrict__ dst, int n)
{
    int idx = blockIdx.x * blockDim.x + threadIdx.x;
    if (idx < n) {
        float r00 = Lr[0] * Lr[0] + Lr[1] * Lr[1] + EPS;
        float r01 = Lr[0] * Lr[2] + Lr[1] * Lr[3];
        float r11 = Lr[2] * Lr[2] + Lr[3] * Lr[3] + EPS;
        v4f v = {r00, r01, r01, r11};
        __builtin_nontemporal_store(v, &dst[idx]);
    }
}

// ---------------------------------------------------------------------------
extern "C" void kernel_launch(void* const* d_in, const int* in_sizes, int n_in,
                              void* d_out, int out_size, void* d_ws,
                              size_t ws_size, hipStream_t stream)
{
    (void)in_sizes; (void)n_in; (void)out_size; (void)ws_size;

    const float* inp = (const float*)d_in[0];  // [B,T,M]
    const float* Fm  = (const float*)d_in[1];  // [S,S]
    const float* Hm  = (const float*)d_in[2];  // [M,S]
    const float* Lq  = (const float*)d_in[3];  // [S,S]
    const float* Lr  = (const float*)d_in[4];  // [M,M]

    float* out  = (float*)d_out;
    float* ws   = (float*)d_ws;
    float* wcov = ws;                 // T*S*S           = 115200 floats
    float* wA   = ws + 115200;        // T*32*24         = 153600 floats
    float* wBT  = ws + 268800;        // T*2*32*2        =  25600 floats

    float* means = out + MEANS_OFF;
    float* covs  = out + COVS_OFF;
    float* Rs    = out + RS_OFF;
    float* Hs    = out + HS_OFF;

    // 1) data-independent covariance / gain sequence (single workgroup)
    kf_cov_seq<<<1, 576, 0, stream>>>(Fm, Hm, Lq, Lr, wcov, wA, wBT);

    // 2) covs broadcast: [B,T,S,S] <- cov_t  (472 MB of NT writes)
    {
        constexpr int per4 = (T_N * S_N * S_N) / 4;   // 28800
        int n4 = (B_N * T_N * S_N * S_N) / 4;         // 29491200
        kf_broadcast4<per4><<<(n4 + 255) / 256, 256, 0, stream>>>(
            (const v4f*)wcov, (v4f*)covs, n4);
    }

    // 3) WMMA mean recurrence: 64 waves, 16 batches each
    kf_means<<<B_N / 16, 32, 0, stream>>>(inp, wA, wBT, means);

    // 4) Rs broadcast: each 128-bit store is one full 2x2 R
    {
        int n = (B_N * T_N * M_N * M_N) / 4;          // 204800
        kf_broadcast_R<<<(n + 255) / 256, 256, 0, stream>>>(Lr, (v4f*)Rs, n);
    }

    // 5) Hs broadcast: [B,T,M,S] <- H (48 floats = 12 x 128-bit period)
    {
        constexpr int per4 = (M_N * S_N) / 4;         // 12
        int n4 = (B_N * T_N * M_N * S_N) / 4;         // 2457600
        kf_broadcast4<per4><<<(n4 + 255) / 256, 256, 0, stream>>>(
            (const v4f*)Hm, (v4f*)Hs, n4);
    }
}